// LlamaAttention_61409442398451
// MI455X (gfx1250) — compile-verified
//
#include <hip/hip_runtime.h>
#include <hip/hip_bf16.h>
#include <math.h>

// ---------------------------------------------------------------------------
// Llama GQA decode step, MI455X (gfx1250, wave32).
//   B=64, HID=4096, NH=32, NKV=8, HD=128, GRP=4, T=4096
// Roofline: 2.1 GB KV stream (~92us @ 23.3TB/s) + 164 MB fp32 weights (~7us).
// GEMMs: native fp32 WMMA (v_wmma_f32_16x16x4_f32), each weight element read
// exactly once (one block covers all 4 M-tiles of M=64). Attention: float4
// coalesced streaming + wave-uniform-branch flash softmax (1 exp/head/key).
// ---------------------------------------------------------------------------

#define HID 4096
#define NH  32
#define NKV 8
#define HD  128
#define GRP 4
#define BB  64
#define TT  4096
#define QKV_N ((NH + 2 * NKV) * HD)   // 6144
#define SCALE 0.08838834764831845f    // 128^-0.5

typedef __attribute__((ext_vector_type(2))) float v2f;
typedef __attribute__((ext_vector_type(8))) float v8f;

// ---------------------------------------------------------------------------
// fp32 GEMM:  C[64, N] = A[64, 4096] @ W[4096, N]  via V_WMMA_F32_16X16X4_F32
// block = 128 threads (4 waves); wave -> one 16-col stripe, ALL 4 M-tiles
// (4 independent accumulators -> W streamed exactly once, WMMA RAW chain has
// 4-instruction spacing). grid = (N/64).
// A frag (16x4, ISA 7.12.2): lanes 0-15 -> K=0,1 ; lanes 16-31 -> K=2,3
// B frag (4x16):             lanes 0-15 -> rows K=0,1 ; lanes 16-31 -> K=2,3
// D (16x16 f32): VGPR j: lanes 0-15 -> M=j, lanes 16-31 -> M=j+8, N=lane&15
// ---------------------------------------------------------------------------
template <int N>
__global__ __launch_bounds__(128)
void gemm_wmma(const float* __restrict__ A, const float* __restrict__ W,
               float* __restrict__ C) {
    const int lane = threadIdx.x & 31;
    const int wave = threadIdx.x >> 5;
    const int n0   = blockIdx.x * 64 + wave * 16;
    const int r    = lane & 15;
    const int koff = (lane < 16) ? 0 : 2;

    const float* __restrict__ arow = A + (size_t)r * HID + koff;
    const float* __restrict__ bcol = W + (size_t)koff * N + (n0 + r);

    v8f c[4] = {{}, {}, {}, {}};
    #pragma unroll 2
    for (int k = 0; k < HID; k += 4) {
        v2f b;
        b.x = bcol[(size_t)k * N];
        b.y = bcol[(size_t)k * N + N];
        #pragma unroll
        for (int mt = 0; mt < 4; ++mt) {
            v2f a;
            a.x = arow[k + mt * 16 * HID];
            a.y = arow[k + 1 + mt * 16 * HID];
            c[mt] = __builtin_amdgcn_wmma_f32_16x16x4_f32(
                        false, a, false, b, (short)0, c[mt], false, false);
        }
    }

    const int row0 = (lane < 16) ? 0 : 8;
    #pragma unroll
    for (int mt = 0; mt < 4; ++mt)
        #pragma unroll
        for (int j = 0; j < 8; ++j)
            C[(size_t)(mt * 16 + row0 + j) * N + n0 + r] = c[mt][j];
}

// ---------------------------------------------------------------------------
// RoPE (NeoX) on fresh q/k + split qkv buffer into q_rot / k_new / v_new.
// grid = (48, B): slot < 32 -> q head; slot < 40 -> k head; else v head copy.
// block = 64 threads (one per rotation pair).
// ---------------------------------------------------------------------------
__global__ __launch_bounds__(64)
void rope_split(const float* __restrict__ qkv, const int* __restrict__ pos,
                float* __restrict__ q_rot, float* __restrict__ k_new,
                float* __restrict__ v_new) {
    const int s = blockIdx.x;   // 0..47
    const int b = blockIdx.y;   // 0..63
    const int d = threadIdx.x;  // 0..63  (half = HD/2)
    const float* src = qkv + (size_t)b * QKV_N;

    if (s >= 40) {              // v head: straight copy
        const int h = s - 40;
        const float* v = src + NH * HD + NKV * HD + h * HD;
        float* dst = v_new + (size_t)(b * NKV + h) * HD;
        dst[d]      = v[d];
        dst[d + 64] = v[d + 64];
        return;
    }
    const int base = (s < 32) ? s * HD : NH * HD + (s - 32) * HD;
    const float x1 = src[base + d];
    const float x2 = src[base + d + 64];
    const float inv_freq = __powf(10000.0f, -(float)d * (1.0f / 64.0f));
    const float fr = (float)pos[b] * inv_freq;
    float sn, cs;
    __sincosf(fr, &sn, &cs);
    const float o1 = x1 * cs - x2 * sn;
    const float o2 = x2 * cs + x1 * sn;
    float* dst = (s < 32) ? (q_rot + (size_t)(b * NH + s) * HD)
                          : (k_new + (size_t)(b * NKV + (s - 32)) * HD);
    dst[d]      = o1;
    dst[d + 64] = o2;
}

// ---------------------------------------------------------------------------
// Flash-decode attention. grid = (NKV, B), block = 256 (8 waves).
// Each wave strides keys t = wave, wave+8, ... over 4097 keys (t==T -> new
// token). Per key: float4 coalesced K/V loads (512B/wave), 4-head dot via
// butterfly shuffles, wave-uniform-branch online softmax (scores are uniform
// across the wave after reduction -> branch is divergence-free, common path
// costs a single v_exp_f32 per head). Cross-wave merge in LDS.
// ---------------------------------------------------------------------------
__global__ __launch_bounds__(256)
void attn(const float* __restrict__ q_rot, const float* __restrict__ k_cache,
          const float* __restrict__ v_cache, const float* __restrict__ k_new,
          const float* __restrict__ v_new, float* __restrict__ ctx) {
    __shared__ float q_s[GRP * HD];            // 2 KB (pre-scaled by SCALE)
    __shared__ float red_m[8][GRP];
    __shared__ float red_l[8][GRP];
    __shared__ float red_acc[8][GRP][HD];      // 16 KB

    const int kv   = blockIdx.x;
    const int b    = blockIdx.y;
    const int tid  = threadIdx.x;
    const int wave = tid >> 5;
    const int lane = tid & 31;

    for (int i = tid; i < GRP * HD; i += 256)
        q_s[i] = q_rot[(size_t)(b * NH + kv * GRP) * HD + i] * SCALE;
    __syncthreads();

    // q fragments for this lane (4 dims per lane per head), hoisted.
    float4 qf[GRP];
    #pragma unroll
    for (int h = 0; h < GRP; ++h)
        qf[h] = *(const float4*)(q_s + h * HD + lane * 4);

    const float* kb = k_cache + ((size_t)b * TT * NKV + kv) * HD;
    const float* vb = v_cache + ((size_t)b * TT * NKV + kv) * HD;
    const float* kn = k_new + (size_t)(b * NKV + kv) * HD + lane * 4;
    const float* vn = v_new + (size_t)(b * NKV + kv) * HD + lane * 4;

    float m[GRP], l[GRP];
    float4 acc[GRP];
    #pragma unroll
    for (int h = 0; h < GRP; ++h) {
        m[h] = -1e30f; l[h] = 0.0f;
        acc[h] = make_float4(0.f, 0.f, 0.f, 0.f);
    }

    for (int t = wave; t <= TT; t += 8) {
        float4 kv4, vv4;
        if (t == TT) { kv4 = *(const float4*)kn; vv4 = *(const float4*)vn; }
        else {
            kv4 = *(const float4*)(kb + (size_t)t * (NKV * HD) + lane * 4);
            vv4 = *(const float4*)(vb + (size_t)t * (NKV * HD) + lane * 4);
        }
        float p[GRP];
        #pragma unroll
        for (int h = 0; h < GRP; ++h)
            p[h] = qf[h].x * kv4.x + qf[h].y * kv4.y +
                   qf[h].z * kv4.z + qf[h].w * kv4.w;
        #pragma unroll
        for (int off = 16; off > 0; off >>= 1) {
            #pragma unroll
            for (int h = 0; h < GRP; ++h)
                p[h] += __shfl_xor(p[h], off, 32);
        }
        // s_h is identical across all 32 lanes -> uniform branch, no
        // divergence; common path (max unchanged) needs one exp only.
        #pragma unroll
        for (int h = 0; h < GRP; ++h) {
            const float sh = p[h];
            if (sh <= m[h]) {
                const float w = __expf(sh - m[h]);
                l[h] += w;
                acc[h].x += w * vv4.x;
                acc[h].y += w * vv4.y;
                acc[h].z += w * vv4.z;
                acc[h].w += w * vv4.w;
            } else {
                const float corr = __expf(m[h] - sh);   // new weight = 1
                l[h] = l[h] * corr + 1.0f;
                acc[h].x = acc[h].x * corr + vv4.x;
                acc[h].y = acc[h].y * corr + vv4.y;
                acc[h].z = acc[h].z * corr + vv4.z;
                acc[h].w = acc[h].w * corr + vv4.w;
                m[h] = sh;
            }
        }
    }

    #pragma unroll
    for (int h = 0; h < GRP; ++h) {
        if (lane == 0) { red_m[wave][h] = m[h]; red_l[wave][h] = l[h]; }
        red_acc[wave][h][lane * 4 + 0] = acc[h].x;
        red_acc[wave][h][lane * 4 + 1] = acc[h].y;
        red_acc[wave][h][lane * 4 + 2] = acc[h].z;
        red_acc[wave][h][lane * 4 + 3] = acc[h].w;
    }
    __syncthreads();

    for (int idx = tid; idx < GRP * HD; idx += 256) {
        const int h = idx >> 7;
        const int d = idx & (HD - 1);
        float M = -1e30f;
        #pragma unroll
        for (int w = 0; w < 8; ++w) M = fmaxf(M, red_m[w][h]);
        float L = 0.f, O = 0.f;
        #pragma unroll
        for (int w = 0; w < 8; ++w) {
            const float e = __expf(red_m[w][h] - M);
            L += red_l[w][h] * e;
            O += red_acc[w][h][d] * e;
        }
        ctx[(size_t)(b * NH + kv * GRP + h) * HD + d] = O / L;
    }
}

// ---------------------------------------------------------------------------
extern "C" void kernel_launch(void* const* d_in, const int* in_sizes, int n_in,
                              void* d_out, int out_size, void* d_ws, size_t ws_size,
                              hipStream_t stream) {
    const int*   positions = (const int*)d_in[0];
    const float* hidden    = (const float*)d_in[1];
    const float* k_cache   = (const float*)d_in[2];
    const float* v_cache   = (const float*)d_in[3];
    const float* w_qkv     = (const float*)d_in[4];
    const float* w_o       = (const float*)d_in[5];
    float* out = (float*)d_out;

    float* ws      = (float*)d_ws;
    float* qkv_buf = ws;                               // 64*6144
    float* q_rot   = qkv_buf + (size_t)BB * QKV_N;     // 64*32*128
    float* k_new   = q_rot + (size_t)BB * NH * HD;     // 64*8*128
    float* v_new   = k_new + (size_t)BB * NKV * HD;    // 64*8*128
    float* ctx     = v_new + (size_t)BB * NKV * HD;    // 64*32*128

    // 1) fused QKV projection: [64,4096] x [4096,6144], W streamed once
    gemm_wmma<QKV_N><<<dim3(QKV_N / 64), 128, 0, stream>>>(hidden, w_qkv, qkv_buf);
    // 2) RoPE + split
    rope_split<<<dim3(NH + 2 * NKV, BB), 64, 0, stream>>>(qkv_buf, positions, q_rot, k_new, v_new);
    // 3) flash-decode attention over 4096 cached + 1 new token
    attn<<<dim3(NKV, BB), 256, 0, stream>>>(q_rot, k_cache, v_cache, k_new, v_new, ctx);
    // 4) output projection: [64,4096] x [4096,4096] -> d_out
    gemm_wmma<HID><<<dim3(HID / 64), 128, 0, stream>>>(ctx, w_o, out);
}